// graphSAGELayer_84782654423297
// MI455X (gfx1250) — compile-verified
//
#include <hip/hip_runtime.h>

typedef float v2f   __attribute__((ext_vector_type(2)));
typedef float v8f   __attribute__((ext_vector_type(8)));
typedef float fvec4 __attribute__((ext_vector_type(4)));

static constexpr int FD      = 128;   // feature dim
static constexpr int TWO_D   = 256;   // concat dim
static constexpr int TILE_M  = 16;    // nodes per workgroup
static constexpr int FEAT_LD = 260;   // padded LDS row stride (words): 16B-aligned, column reads conflict-free

__global__ __launch_bounds__(128, 2)
void sage_fused_wmma(const float* __restrict__ h,
                     const float* __restrict__ W,
                     const int*   __restrict__ nbr,
                     float*       __restrict__ out,
                     int N, int K, float invK)
{
    __shared__ float feat[TILE_M * FEAT_LD];

    const int tid = threadIdx.x;
    const int m0  = blockIdx.x * TILE_M;

    // ---------------- Phase 1: build feat tile [16 x 256] in LDS ----------------
    // 128 threads: 8 threads per node row, each owns 16 consecutive floats.
    {
        const int row  = tid >> 3;      // 0..15
        const int seg  = tid & 7;       // 0..7
        const int col0 = seg * 16;      // 0,16,...,112
        int g = m0 + row;
        if (g >= N) g = N - 1;          // clamp; duplicate work masked at store

        // self features -> feat[row][0:128]
        const fvec4* hp = reinterpret_cast<const fvec4*>(h + (size_t)g * FD + col0);
        fvec4 s0 = hp[0], s1 = hp[1], s2 = hp[2], s3 = hp[3];
        fvec4* frow = reinterpret_cast<fvec4*>(&feat[row * FEAT_LD + col0]);
        frow[0] = s0; frow[1] = s1; frow[2] = s2; frow[3] = s3;

        // neighbor mean -> feat[row][128:256]
        fvec4 a0 = {}, a1 = {}, a2 = {}, a3 = {};
        const int* nrow = nbr + (size_t)g * K;
        for (int k = 0; k < K; ++k) {
            const int idx = nrow[k];
            const fvec4* np = reinterpret_cast<const fvec4*>(h + (size_t)idx * FD + col0);
            a0 += np[0]; a1 += np[1]; a2 += np[2]; a3 += np[3];
        }
        fvec4* mrow = reinterpret_cast<fvec4*>(&feat[row * FEAT_LD + FD + col0]);
        mrow[0] = a0 * invK; mrow[1] = a1 * invK; mrow[2] = a2 * invK; mrow[3] = a3 * invK;
    }

    __syncthreads();

    // ---------------- Phase 2: out[16 x 128] = feat[16 x 256] @ W[256 x 128] ----------------
    // 4 waves; wave w owns output columns [w*32, w*32+32) as two 16x16 WMMA N-tiles.
    const int lane = tid & 31;
    const int wave = tid >> 5;
    const int m    = lane & 15;     // A/M row index (per lane half); N column within tile for B/C
    const int kh   = lane >> 4;     // 0: K pair {k,k+1}; 1: K pair {k+2,k+3}  (16x4 f32 A layout)
    const int n0   = wave * 32;

    v8f c0 = {};
    v8f c1 = {};

    const float* wp = W + n0 + m;   // column base for this lane

    #pragma unroll 8
    for (int k0 = 0; k0 < TWO_D; k0 += 4) {
        const int kr = k0 + 2 * kh;

        // A fragment: 16x4 f32, one K-pair per lane half (8B-aligned LDS read)
        v2f a = *reinterpret_cast<const v2f*>(&feat[m * FEAT_LD + kr]);

        // B fragments: 4x16 f32 tiles of W (row-major, L2/WGP$-resident)
        v2f b0, b1;
        b0.x = wp[(size_t)kr * FD];
        b0.y = wp[(size_t)(kr + 1) * FD];
        b1.x = wp[(size_t)kr * FD + 16];
        b1.y = wp[(size_t)(kr + 1) * FD + 16];

        c0 = __builtin_amdgcn_wmma_f32_16x16x4_f32(false, a, false, b0, (short)0, c0, false, false);
        c1 = __builtin_amdgcn_wmma_f32_16x16x4_f32(false, a, false, b1, (short)0, c1, false, false);
    }

    // C/D layout: VGPR r -> M = r + 8*(lane/16), N = lane%16
    float* orow = out + (size_t)(m0 + 8 * kh) * FD + n0 + m;
    if (m0 + TILE_M <= N) {
        // Fast path (all full tiles when N % 16 == 0): straight-line, clause-able stores.
        #pragma unroll
        for (int r = 0; r < 8; ++r) {
            orow[(size_t)r * FD]      = c0[r];
            orow[(size_t)r * FD + 16] = c1[r];
        }
    } else {
        #pragma unroll
        for (int r = 0; r < 8; ++r) {
            if (m0 + r + 8 * kh < N) {
                orow[(size_t)r * FD]      = c0[r];
                orow[(size_t)r * FD + 16] = c1[r];
            }
        }
    }
}

extern "C" void kernel_launch(void* const* d_in, const int* in_sizes, int n_in,
                              void* d_out, int out_size, void* d_ws, size_t ws_size,
                              hipStream_t stream) {
    const float* h   = (const float*)d_in[0];
    const float* W   = (const float*)d_in[1];
    const int*   nbr = (const int*)d_in[2];
    float*       out = (float*)d_out;

    const int N = in_sizes[0] / FD;               // 50000
    const int K = in_sizes[2] / (N > 0 ? N : 1);  // 25
    const float invK = 1.0f / (float)K;

    const int grid = (N + TILE_M - 1) / TILE_M;   // 3125
    sage_fused_wmma<<<grid, 128, 0, stream>>>(h, W, nbr, out, N, K, invK);
}